// Stackgram_56178172231965
// MI455X (gfx1250) — compile-verified
//
#include <hip/hip_runtime.h>
#include <cstdint>

// Problem constants (from the reference): x is (B,1,H,T) f32, out is (B,T,H,W) f32.
constexpr int B_ = 2;
constexpr int T_ = 180;
constexpr int H_ = 512;
constexpr int W_ = 512;

constexpr int NTHREADS = 256;        // 8 wave32 waves per block
constexpr int ROWS_PER_BLOCK = 64;   // each block: one (b,t) line, 64 output rows

typedef float v4f __attribute__((ext_vector_type(4)));

// out[b,t,h,w] = valid ? x[b,0,iy,t] : 0
//   fx = sin*256*xs + cos*256*ys + 255.5 ; ix = rndne(fx)   (validity only)
//   fy = cos*256*xs - sin*256*ys + 255.5 ; iy = rndne(fy)   (gather index)
//   xs = (w+0.5)/256 - 1 ; ys = (h+0.5)/256 - 1
__global__ __launch_bounds__(NTHREADS)
void stackgram_kernel(const float* __restrict__ x, float* __restrict__ out) {
    const int t = blockIdx.x;            // 0..179
    const int b = blockIdx.z;            // 0..1
    const int row0 = blockIdx.y * ROWS_PER_BLOCK;
    const int tid = (int)threadIdx.x;

    __shared__ float line[H_];

    // ---- Stage the strided column x[b,0,:,t] (stride T_ floats) into LDS via
    // CDNA5 async global->LDS loads (ASYNCcnt). Per-lane global byte-offset
    // (GVS mode) and per-lane LDS byte-address.
    {
        const uint32_t lds_base = (uint32_t)(uintptr_t)&line[0];
        const uint64_t gbase = (uint64_t)(uintptr_t)x;
#pragma unroll
        for (int i = 0; i < H_; i += NTHREADS) {
            const int h = i + tid;
            const uint32_t ldsa = lds_base + (uint32_t)h * 4u;
            const uint32_t goff = (uint32_t)(((b * H_ + h) * T_ + t) * 4);
            asm volatile("global_load_async_to_lds_b32 %0, %1, %2"
                         :
                         : "v"(ldsa), "v"(goff), "s"(gbase)
                         : "memory");
        }
        asm volatile("s_wait_asynccnt 0x0" ::: "memory");
    }
    __syncthreads();

    // Per-block uniform trig (t uniform -> scalarized, cold code).
    const float th = (float)t * 0.017453292519943295f; // deg2rad
    const float sn = sinf(th);
    const float cs = cosf(th);
    const float s256 = sn * 256.0f;
    const float c256 = cs * 256.0f;

    // Thread -> (row parity, 4-wide w quad). 128 threads cover one 512-wide row.
    const int w0 = (tid & 127) << 2;
    const int rpar = tid >> 7; // 0 or 1
    const float xs0 = ((float)w0 + 0.5f) * (1.0f / 256.0f) - 1.0f;

    float* dst0 = out + ((((size_t)b * T_ + t) * H_) * W_ + (size_t)w0);

#pragma unroll 2
    for (int h = row0 + rpar; h < row0 + ROWS_PER_BLOCK; h += 2) {
        const float ys = ((float)h + 0.5f) * (1.0f / 256.0f) - 1.0f;
        // Per-row bases; per-element step along w is exactly (sn, cs).
        const float fx0 = fmaf(s256, xs0, fmaf(c256, ys, 255.5f));
        const float fy0 = fmaf(c256, xs0, fmaf(-s256, ys, 255.5f));

        v4f o;
#pragma unroll
        for (int k = 0; k < 4; ++k) {
            const float fx = fmaf((float)k, sn, fx0);
            const float fy = fmaf((float)k, cs, fy0);
            // round-half-to-even matches jnp.round
            const int ix = (int)rintf(fx);
            const int iy = (int)rintf(fy);
            // both in [0,512) <=> (ix|iy) unsigned < 512
            const bool valid = ((unsigned)(ix | iy)) < (unsigned)W_;
            const float v = line[iy & (H_ - 1)];     // ds_load_b32 (safe addr)
            o[k] = valid ? v : 0.0f;
        }

        // Coalesced 16B non-temporal store (write-once output >> L2).
        __builtin_nontemporal_store(o, (v4f*)(dst0 + (size_t)h * W_));
    }
}

extern "C" void kernel_launch(void* const* d_in, const int* in_sizes, int n_in,
                              void* d_out, int out_size, void* d_ws, size_t ws_size,
                              hipStream_t stream) {
    (void)in_sizes; (void)n_in; (void)d_ws; (void)ws_size; (void)out_size;
    const float* x = (const float*)d_in[0];
    float* out = (float*)d_out;

    dim3 grid(T_, H_ / ROWS_PER_BLOCK, B_); // (180, 8, 2)
    dim3 block(NTHREADS);
    stackgram_kernel<<<grid, block, 0, stream>>>(x, out);
}